// CluTSP_Encoder_75136157876543
// MI455X (gfx1250) — compile-verified
//
#include <hip/hip_runtime.h>
#include <math.h>

// ---------------------------------------------------------------------------
// CDNA5 (gfx1250) implementation of the CluTSP GATv2 encoder.
// GEMMs use V_WMMA_F32_16X16X4_F32 (full fp32 matrix path; GEMMs here are
// HBM-bound at AI~32 FLOP/B so fp32 costs nothing vs fp16 and matches the
// f32 reference numerics). Edge softmax/aggregation = wave-per-edge with
// float4 coalesced channel reads and L2 float atomics.
// ---------------------------------------------------------------------------

typedef float v2f __attribute__((ext_vector_type(2)));
typedef float v8f __attribute__((ext_vector_type(8)));

// ---------------- WMMA GEMM: C[M x Ncol] = A[M x K] @ W[K x Ncol] (+bias)(+elu)(+=)
// One wave computes a 16-row x 128-col strip (8 WMMA tiles, v8f accum each).
// FLAGS: 1=bias, 2=ELU, 4=accumulate into C.
template <int FLAGS>
__global__ __launch_bounds__(256) void gemm_wmma_k(
    const float* __restrict__ A, const float* __restrict__ W,
    const float* __restrict__ bias, float* __restrict__ C,
    int M, int K, int Ncol) {
  const int rowTiles = (M + 15) >> 4;
  const int colBlks = Ncol >> 7;  // 128-column blocks
  int gw = (int)((blockIdx.x * blockDim.x + threadIdx.x) >> 5);
  if (gw >= rowTiles * colBlks) return;  // uniform per wave -> EXEC all ones
  int rowTile = gw % rowTiles;
  int colBlk = gw / rowTiles;
  int lane = threadIdx.x & 31;
  int khalf = lane >> 4;  // ISA: lanes 16-31 hold K+2,K+3 of each 4-step
  int l15 = lane & 15;
  int arow = rowTile * 16 + l15;
  if (arow >= M) arow = M - 1;  // clamp; padded rows never stored
  const float* Arow = A + (size_t)arow * K;
  int gcolBase = colBlk * 128 + l15;

  v8f acc[8] = {};
  for (int k = 0; k < K; k += 4) {
    v2f a = *(const v2f*)(Arow + k + 2 * khalf);
#pragma unroll
    for (int t = 0; t < 8; ++t) {
      int gcol = gcolBase + t * 16;
      v2f b;
      b.x = W[(size_t)(k + 2 * khalf) * Ncol + gcol];
      b.y = W[(size_t)(k + 2 * khalf + 1) * Ncol + gcol];
      acc[t] = __builtin_amdgcn_wmma_f32_16x16x4_f32(
          false, a, false, b, (short)0, acc[t], false, false);
    }
  }

  int rowBase = rowTile * 16 + khalf * 8;  // C layout: vgpr i = row 8*half+i
#pragma unroll
  for (int t = 0; t < 8; ++t) {
    int gcol = gcolBase + t * 16;
    float bv = (FLAGS & 1) ? bias[gcol] : 0.0f;
#pragma unroll
    for (int i = 0; i < 8; ++i) {
      int r = rowBase + i;
      if (r < M) {
        float v = acc[t][i] + bv;
        if (FLAGS & 2) v = (v > 0.0f) ? v : expm1f(v);  // ELU
        size_t idx = (size_t)r * Ncol + gcol;
        if (FLAGS & 4) C[idx] += v; else C[idx] = v;
      }
    }
  }
}

// ---------------- helpers: monotone float<->uint for atomic float max -------
__device__ __forceinline__ unsigned f2mono(float f) {
  unsigned u = __float_as_uint(f);
  return (u & 0x80000000u) ? ~u : (u | 0x80000000u);
}
__device__ __forceinline__ float mono2f(unsigned u) {
  return __uint_as_float((u & 0x80000000u) ? (u & 0x7FFFFFFFu) : ~u);
}

// ---------------- edge pass 1: alpha[e,h] + segment max over dst ------------
// One wave per edge; lane owns 4 channels (float4), head = lane>>2.
template <int HAS_E>
__global__ __launch_bounds__(256) void edge_alpha_k(
    const float* __restrict__ xl, const float* __restrict__ xr,
    const int* __restrict__ src, const int* __restrict__ dst,
    const float* __restrict__ eattr, const float* __restrict__ easum,
    float invE, const float* __restrict__ We, const float* __restrict__ att,
    float* __restrict__ alpha, unsigned* __restrict__ amaxm, int E, int Etot) {
  int gw = (int)((blockIdx.x * blockDim.x + threadIdx.x) >> 5);
  if (gw >= Etot) return;
  int lane = threadIdx.x & 31;
  int e = gw, s, d;
  float w = 0.0f;
  if (e < E) { s = src[e]; d = dst[e]; if (HAS_E) w = eattr[e]; }
  else       { s = d = e - E;          if (HAS_E) w = easum[0] * invE; }
  float4 vl = *(const float4*)(xl + (size_t)s * 128 + lane * 4);
  float4 vr = *(const float4*)(xr + (size_t)d * 128 + lane * 4);
  float mx = vl.x + vr.x, my = vl.y + vr.y, mz = vl.z + vr.z, mw = vl.w + vr.w;
  if (HAS_E) {
    float4 we = *(const float4*)(We + lane * 4);
    mx += w * we.x; my += w * we.y; mz += w * we.z; mw += w * we.w;
  }
  float4 at = *(const float4*)(att + lane * 4);
  float p = at.x * (mx > 0.f ? mx : 0.2f * mx)
          + at.y * (my > 0.f ? my : 0.2f * my)
          + at.z * (mz > 0.f ? mz : 0.2f * mz)
          + at.w * (mw > 0.f ? mw : 0.2f * mw);
  p += __shfl_xor(p, 1, 32);
  p += __shfl_xor(p, 2, 32);
  if ((lane & 3) == 0) {
    int h = lane >> 2;
    alpha[(size_t)e * 8 + h] = p;
    atomicMax(&amaxm[(size_t)d * 8 + h], f2mono(p));
  }
}

// ---------------- edge pass 2: ex = exp(alpha - amax[dst]); den += ex -------
__global__ __launch_bounds__(256) void edge_expsum_k(
    const int* __restrict__ dst, float* __restrict__ alpha,
    const unsigned* __restrict__ amaxm, float* __restrict__ den,
    int E, int Etot) {
  int idx = blockIdx.x * blockDim.x + threadIdx.x;
  if (idx >= Etot * 8) return;
  int e = idx >> 3, h = idx & 7;
  int d = (e < E) ? dst[e] : (e - E);
  float am = mono2f(amaxm[(size_t)d * 8 + h]);
  float ex = expf(alpha[idx] - am);
  alpha[idx] = ex;
  atomicAdd(&den[(size_t)d * 8 + h], ex);
}

// ---------------- edge pass 3: out[dst] += xl[src] * ex/(den+1e-16) ---------
__global__ __launch_bounds__(256) void edge_agg_k(
    const float* __restrict__ xl, const int* __restrict__ src,
    const int* __restrict__ dst, const float* __restrict__ alpha,
    const float* __restrict__ den, float* __restrict__ outp, int E, int Etot) {
  int gw = (int)((blockIdx.x * blockDim.x + threadIdx.x) >> 5);
  if (gw >= Etot) return;
  int lane = threadIdx.x & 31;
  int e = gw, s, d;
  if (e < E) { s = src[e]; d = dst[e]; } else { s = d = e - E; }
  int h = lane >> 2;
  float a = alpha[(size_t)e * 8 + h] / (den[(size_t)d * 8 + h] + 1e-16f);
  float4 vl = *(const float4*)(xl + (size_t)s * 128 + lane * 4);
  float* o = outp + (size_t)d * 128 + lane * 4;
  atomicAdd(o + 0, vl.x * a);
  atomicAdd(o + 1, vl.y * a);
  atomicAdd(o + 2, vl.z * a);
  atomicAdd(o + 3, vl.w * a);
}

// ---------------- misc small kernels ----------------------------------------
__global__ void init_bias_k(float* __restrict__ o, const float* __restrict__ b, int n) {
  int i = blockIdx.x * blockDim.x + threadIdx.x;
  if (i < n) o[i] = b[i & 127];
}

__global__ void embed_k(const float* __restrict__ x, const float* __restrict__ W0,
                        const float* __restrict__ b0, float* __restrict__ emb, int N) {
  int i = blockIdx.x * blockDim.x + threadIdx.x;
  if (i >= N * 128) return;
  int n = i >> 7, c = i & 127;
  emb[i] = x[n * 2] * W0[c] + x[n * 2 + 1] * W0[128 + c] + b0[c];
}

__global__ void reduce_sum_k(const float* __restrict__ a, int n, float* __restrict__ out) {
  __shared__ float sm[256];
  float s = 0.f;
  for (int i = blockIdx.x * blockDim.x + threadIdx.x; i < n; i += gridDim.x * blockDim.x)
    s += a[i];
  sm[threadIdx.x] = s;
  __syncthreads();
  for (int st = 128; st > 0; st >>= 1) {
    if ((int)threadIdx.x < st) sm[threadIdx.x] += sm[threadIdx.x + st];
    __syncthreads();
  }
  if (threadIdx.x == 0) atomicAdd(out, sm[0]);
}

// s = a + b; per-column sum & sumsq (thread = column, block strides rows)
__global__ __launch_bounds__(128) void bn_stats_k(
    const float* __restrict__ a, const float* __restrict__ b, float* __restrict__ s,
    float* __restrict__ colsum, float* __restrict__ colsq, int N) {
  int c = threadIdx.x;
  float acc = 0.f, acc2 = 0.f;
  for (int r = blockIdx.x; r < N; r += gridDim.x) {
    size_t i = (size_t)r * 128 + c;
    float v = a[i] + b[i];
    s[i] = v;
    acc += v;
    acc2 += v * v;
  }
  atomicAdd(&colsum[c], acc);
  atomicAdd(&colsq[c], acc2);
}

__global__ void bn_apply_k(const float* __restrict__ s, float* __restrict__ y,
                           const float* __restrict__ colsum, const float* __restrict__ colsq,
                           const float* __restrict__ g, const float* __restrict__ b, int N) {
  int i = blockIdx.x * blockDim.x + threadIdx.x;
  if (i >= N * 128) return;
  int c = i & 127;
  float inv = 1.0f / (float)N;
  float mu = colsum[c] * inv;
  float var = colsq[c] * inv - mu * mu;
  y[i] = (s[i] - mu) * (1.0f / sqrtf(var + 1e-5f)) * g[c] + b[c];
}

// ---------------- host orchestration ----------------------------------------
namespace {

struct Ctx {
  float *emb, *xl, *xr, *attb, *x1, *x2, *big, *alpha, *den, *colsum, *colsq, *easum;
  unsigned* amax;
  int N;
  hipStream_t st;
};

struct BranchP {
  const float *Wl, *bl, *Wr, *br, *att, *bias, *We;
  const float *mhcW, *mhcb, *bn1g, *bn1b, *ff1W, *ff1b, *ff2W, *ff2b, *bn2g, *bn2b;
};

void gemm(const Ctx& c, int flags, float* C, const float* A, const float* W,
          const float* bias, int M, int K, int Ncol) {
  int waves = ((M + 15) / 16) * (Ncol / 128);
  int blocks = (waves + 7) / 8;
  dim3 g(blocks), b(256);
  switch (flags) {
    case 1: gemm_wmma_k<1><<<g, b, 0, c.st>>>(A, W, bias, C, M, K, Ncol); break;
    case 3: gemm_wmma_k<3><<<g, b, 0, c.st>>>(A, W, bias, C, M, K, Ncol); break;
    case 4: gemm_wmma_k<4><<<g, b, 0, c.st>>>(A, W, bias, C, M, K, Ncol); break;
    default: gemm_wmma_k<0><<<g, b, 0, c.st>>>(A, W, bias, C, M, K, Ncol); break;
  }
}

void run_branch(const Ctx& c, const BranchP& P, const int* src, const int* dst,
                const float* eattr, int E, int hasE, float* xout) {
  const int N = c.N, Etot = E + N;
  gemm(c, 1, c.xl, c.emb, P.Wl, P.bl, N, 128, 128);  // lin_l
  gemm(c, 1, c.xr, c.emb, P.Wr, P.br, N, 128, 128);  // lin_r
  hipMemsetAsync(c.amax, 0, (size_t)N * 8 * 4, c.st);
  hipMemsetAsync(c.den, 0, (size_t)N * 8 * 4, c.st);
  int eb = (Etot + 7) / 8;
  float invE = 1.0f / (float)E;
  if (hasE)
    edge_alpha_k<1><<<eb, 256, 0, c.st>>>(c.xl, c.xr, src, dst, eattr, c.easum,
                                          invE, P.We, P.att, c.alpha, c.amax, E, Etot);
  else
    edge_alpha_k<0><<<eb, 256, 0, c.st>>>(c.xl, c.xr, src, dst, eattr, c.easum,
                                          invE, P.We, P.att, c.alpha, c.amax, E, Etot);
  edge_expsum_k<<<(Etot * 8 + 255) / 256, 256, 0, c.st>>>(dst, c.alpha, c.amax, c.den, E, Etot);
  init_bias_k<<<(N * 128 + 255) / 256, 256, 0, c.st>>>(c.attb, P.bias, N * 128);
  edge_agg_k<<<eb, 256, 0, c.st>>>(c.xl, src, dst, c.alpha, c.den, c.attb, E, Etot);
  gemm(c, 1, c.xl, c.attb, P.mhcW, P.mhcb, N, 128, 128);  // multi-head combine
  hipMemsetAsync(c.colsum, 0, 512, c.st);
  hipMemsetAsync(c.colsq, 0, 512, c.st);
  bn_stats_k<<<512, 128, 0, c.st>>>(c.xl, c.emb, c.xr, c.colsum, c.colsq, N);  // h+res
  bn_apply_k<<<(N * 128 + 255) / 256, 256, 0, c.st>>>(c.xr, c.attb, c.colsum, c.colsq,
                                                      P.bn1g, P.bn1b, N);  // gx
  gemm(c, 3, c.big, c.attb, P.ff1W, P.ff1b, N, 128, 512);  // elu(ff1)
  gemm(c, 1, c.xl, c.big, P.ff2W, P.ff2b, N, 512, 128);    // ff2
  hipMemsetAsync(c.colsum, 0, 512, c.st);
  hipMemsetAsync(c.colsq, 0, 512, c.st);
  bn_stats_k<<<512, 128, 0, c.st>>>(c.xl, c.attb, c.xr, c.colsum, c.colsq, N);  // h+gx
  bn_apply_k<<<(N * 128 + 255) / 256, 256, 0, c.st>>>(c.xr, xout, c.colsum, c.colsq,
                                                      P.bn2g, P.bn2b, N);
}

}  // namespace

extern "C" void kernel_launch(void* const* d_in, const int* in_sizes, int n_in,
                              void* d_out, int out_size, void* d_ws, size_t ws_size,
                              hipStream_t stream) {
  if (n_in < 44) return;
  const float* x = (const float*)d_in[0];
  const int N = in_sizes[0] / 2;
  const int* ei = (const int*)d_in[1];
  const int E = in_sizes[2];
  const float* ea = (const float*)d_in[2];
  const int* cei = (const int*)d_in[4];
  const int EC = in_sizes[4] / 2;
  const float* W0 = (const float*)d_in[7];
  const float* b0 = (const float*)d_in[8];

  // Detect pytree flattening order of params['layers']:
  // alphabetical (jax tree_leaves): index 19 = cat_W (L*256*128) > index 20 = cat_b.
  // insertion order: index 19 = bn1_b (L*128) < index 20 = ff1_W.
  const bool sorted_leaves = in_sizes[19] > in_sizes[20];
#define FP(i) ((const float*)d_in[i])
  const float *g_Wl, *g_bl, *g_Wr, *g_br, *g_att, *g_We, *g_bias, *mhc_W, *mhc_b,
      *bn1_g, *bn1_b, *ff1_W, *ff1_b, *ff2_W, *ff2_b, *bn2_g, *bn2_b, *c_Wl, *c_bl,
      *c_Wr, *c_br, *c_att, *c_bias, *cmhc_W, *cmhc_b, *cbn1_g, *cbn1_b, *cff1_W,
      *cff1_b, *cff2_W, *cff2_b, *cbn2_g, *cbn2_b, *cat_W, *cat_b;
  int L;
  if (sorted_leaves) {
    bn1_b = FP(9);  bn1_g = FP(10); bn2_b = FP(11); bn2_g = FP(12);
    c_Wl = FP(13);  c_Wr = FP(14);  c_att = FP(15); c_bias = FP(16);
    c_bl = FP(17);  c_br = FP(18);  cat_W = FP(19); cat_b = FP(20);
    cbn1_b = FP(21); cbn1_g = FP(22); cbn2_b = FP(23); cbn2_g = FP(24);
    cff1_W = FP(25); cff1_b = FP(26); cff2_W = FP(27); cff2_b = FP(28);
    cmhc_W = FP(29); cmhc_b = FP(30);
    ff1_W = FP(31); ff1_b = FP(32); ff2_W = FP(33); ff2_b = FP(34);
    g_We = FP(35); g_Wl = FP(36); g_Wr = FP(37); g_att = FP(38);
    g_bias = FP(39); g_bl = FP(40); g_br = FP(41);
    mhc_W = FP(42); mhc_b = FP(43);
    L = in_sizes[20] / 128;
  } else {
    g_Wl = FP(9);  g_bl = FP(10); g_Wr = FP(11); g_br = FP(12);
    g_att = FP(13); g_We = FP(14); g_bias = FP(15);
    mhc_W = FP(16); mhc_b = FP(17); bn1_g = FP(18); bn1_b = FP(19);
    ff1_W = FP(20); ff1_b = FP(21); ff2_W = FP(22); ff2_b = FP(23);
    bn2_g = FP(24); bn2_b = FP(25);
    c_Wl = FP(26); c_bl = FP(27); c_Wr = FP(28); c_br = FP(29);
    c_att = FP(30); c_bias = FP(31);
    cmhc_W = FP(32); cmhc_b = FP(33); cbn1_g = FP(34); cbn1_b = FP(35);
    cff1_W = FP(36); cff1_b = FP(37); cff2_W = FP(38); cff2_b = FP(39);
    cbn2_g = FP(40); cbn2_b = FP(41); cat_W = FP(42); cat_b = FP(43);
    L = in_sizes[43] / 128;
  }
#undef FP

  // ---- workspace carve (256B aligned) ----
  char* ws = (char*)d_ws;
  size_t off = 0;
  auto take = [&](size_t bytes) -> void* {
    void* p = ws + off;
    off += (bytes + 255) & ~((size_t)255);
    return p;
  };
  const size_t NB = (size_t)N * 128 * 4;
  Ctx c;
  c.N = N;
  c.st = stream;
  c.emb = (float*)take(NB);
  c.xl = (float*)take(NB);
  c.xr = (float*)take(NB);
  c.attb = (float*)take(NB);
  c.x1 = (float*)take(NB);
  c.x2 = (float*)take(NB);
  c.big = (float*)take((size_t)N * 512 * 4);
  c.alpha = (float*)take((size_t)(E + N) * 8 * 4);
  c.amax = (unsigned*)take((size_t)N * 8 * 4);
  c.den = (float*)take((size_t)N * 8 * 4);
  c.colsum = (float*)take(512);
  c.colsq = (float*)take(512);
  c.easum = (float*)take(256);
  (void)ws_size;

  // mean(edge_attr) once
  hipMemsetAsync(c.easum, 0, 4, stream);
  reduce_sum_k<<<256, 256, 0, stream>>>(ea, E, c.easum);

  // initial embedding: x(N,2) @ W0(2,128) + b0
  embed_k<<<(N * 128 + 255) / 256, 256, 0, stream>>>(x, W0, b0, c.emb, N);

  for (int l = 0; l < L; ++l) {
    BranchP Pd = {g_Wl + (size_t)l * 16384, g_bl + l * 128, g_Wr + (size_t)l * 16384,
                  g_br + l * 128, g_att + l * 128, g_bias + l * 128, g_We + l * 128,
                  mhc_W + (size_t)l * 16384, mhc_b + l * 128, bn1_g + l * 128,
                  bn1_b + l * 128, ff1_W + (size_t)l * 65536, ff1_b + l * 512,
                  ff2_W + (size_t)l * 65536, ff2_b + l * 128, bn2_g + l * 128,
                  bn2_b + l * 128};
    BranchP Pc = {c_Wl + (size_t)l * 16384, c_bl + l * 128, c_Wr + (size_t)l * 16384,
                  c_br + l * 128, c_att + l * 128, c_bias + l * 128, nullptr,
                  cmhc_W + (size_t)l * 16384, cmhc_b + l * 128, cbn1_g + l * 128,
                  cbn1_b + l * 128, cff1_W + (size_t)l * 65536, cff1_b + l * 512,
                  cff2_W + (size_t)l * 65536, cff2_b + l * 128, cbn2_g + l * 128,
                  cbn2_b + l * 128};

    run_branch(c, Pd, ei, ei + E, ea, E, 1, c.x1);        // distance graph (edge feats)
    run_branch(c, Pc, cei, cei + EC, nullptr, EC, 0, c.x2);  // cluster graph

    // emb_next = [x1|x2] @ cat_W + cat_b  (two GEMMs: write then accumulate)
    float* o = (l == L - 1) ? (float*)d_out : c.emb;
    const float* cw = cat_W + (size_t)l * 32768;
    gemm(c, 1, o, c.x1, cw, cat_b + l * 128, N, 128, 128);
    gemm(c, 4, o, c.x2, cw + 16384, nullptr, N, 128, 128);
  }
  (void)out_size;
}